// EMA_35648228557513
// MI455X (gfx1250) — compile-verified
//
#include <hip/hip_runtime.h>
#include <math.h>
#include <stdint.h>

// EMA-with-SD scan, chunked parallel decomposition for MI455X (gfx1250).
//   T=8192 steps, state (N=32, D=128). Output (T, 2N, D) + final states.
// Memory-bound: ~268 MB stores -> ~12us floor @ 23.3 TB/s. No WMMA (elementwise
// recurrence, per-element coefficients). CDNA5 paths used: async global->LDS
// staging (ASYNCcnt) and non-temporal B128 streaming stores.

typedef float __attribute__((ext_vector_type(4))) f32x4;

#define T_STEPS 8192
#define DIM     128
#define NH      32
#define CHUNK   64                      // L: steps per chunk
#define NCHUNK  (T_STEPS / CHUNK)       // 128 chunks
#define STATE   (NH * DIM)              // 4096 state elements
#define ROW     (2 * NH * DIM)          // 8192 floats per output time-row
#define BLOCK_X 32                      // d-groups of 4 (covers D=128)
#define BLOCK_Y 8                       // heads per block
#define BLOCK_T (BLOCK_X * BLOCK_Y)     // 256 threads = 8 waves (wave32)

#if defined(__HIP_DEVICE_COMPILE__) && \
    __has_builtin(__builtin_amdgcn_global_load_async_to_lds_b128) && \
    __has_builtin(__builtin_amdgcn_s_wait_asynccnt)
#define USE_ASYNC_LDS 1
#else
#define USE_ASYNC_LDS 0
#endif

#if USE_ASYNC_LDS
// The builtin's params are int4-vector pointers in the global (__device__, AS1)
// and LDS (__shared__, AS3) address spaces (per hipcc's own diagnostic).
typedef int v4i __attribute__((vector_size(16)));
typedef __attribute__((address_space(1))) v4i* gbl_v4i_p;
typedef __attribute__((address_space(3))) v4i* lds_v4i_p;

// Straight-line async staging: the b128 async-to-LDS instruction's immediate
// offset applies to BOTH the global and LDS addresses (ISA §10), and both
// advance by BLOCK_T*16 bytes per step, so one VGPR address pair + 8 immediate
// offsets covers the whole 32 KB tile with zero per-iteration address math.
template <int J, int ITERS>
__device__ __forceinline__ void stage_async(uintptr_t g, uint32_t l)
{
    __builtin_amdgcn_global_load_async_to_lds_b128(
        (gbl_v4i_p)g, (lds_v4i_p)l, J * (BLOCK_T * 16), 0);
    if constexpr (J + 1 < ITERS) stage_async<J + 1, ITERS>(g, l);
}
#endif

// Stage one x-chunk (CHUNK x DIM floats = 32 KB) into LDS.
__device__ __forceinline__ void stage_chunk(const float* __restrict__ xg,
                                            float* __restrict__ sx, int tid)
{
    constexpr int NV4   = CHUNK * DIM / 4;   // 2048 16-byte packets
    constexpr int ITERS = NV4 / BLOCK_T;     // 8 packets per thread
#if USE_ASYNC_LDS
    const uintptr_t g = (uintptr_t)xg + (uintptr_t)tid * 16u;
    const uint32_t  l = (uint32_t)(uintptr_t)sx + (uint32_t)tid * 16u;
    stage_async<0, ITERS>(g, l);
    __builtin_amdgcn_s_wait_asynccnt(0);
#else
    #pragma unroll
    for (int j = 0; j < ITERS; ++j) {
        const int i = tid + j * BLOCK_T;
        ((f32x4*)sx)[i] = ((const f32x4*)xg)[i];
    }
#endif
    __syncthreads();
}

// Phase 1: per-chunk zero-start aggregates.
//   q  : zero-start mean scan end  (h_a_end = b^L * H + q)
//   av : zero-start var scan end
//   B  : -2*a*b^L * sum(u_k)       (h_v_end = b^L*Hv + av + B*H + C*H^2)
__global__ __launch_bounds__(BLOCK_T) void ema_phase1(
    const float* __restrict__ x, const float* __restrict__ alpha,
    float* __restrict__ Q, float* __restrict__ A, float* __restrict__ B)
{
    __shared__ __align__(16) float sx[CHUNK * DIM];
    const int c   = blockIdx.x;
    const int tx  = threadIdx.x, ty = threadIdx.y;
    const int tid = ty * BLOCK_X + tx;
    stage_chunk(x + (size_t)c * CHUNK * DIM, sx, tid);

    const int n = blockIdx.y * BLOCK_Y + ty;
    const int i = n * DIM + tx * 4;
    const f32x4 a = *(const f32x4*)(alpha + i);
    const f32x4 b = 1.0f - a;

    f32x4 q  = {0.0f, 0.0f, 0.0f, 0.0f};
    f32x4 av = {0.0f, 0.0f, 0.0f, 0.0f};
    f32x4 su = {0.0f, 0.0f, 0.0f, 0.0f};
    #pragma unroll 8
    for (int k = 0; k < CHUNK; ++k) {
        const f32x4 xk = *(const f32x4*)(sx + k * DIM + tx * 4);
        const f32x4 u  = xk - q;           // x_k - s_k (zero-start state)
        av = b * (av + a * u * u);
        su += u;
        q  += a * u;
    }
    f32x4 bl = b;
    #pragma unroll
    for (int s = 0; s < 6; ++s) bl *= bl;  // b^64 == b^CHUNK

    const size_t o = (size_t)c * STATE + i;
    *(f32x4*)(Q + o) = q;
    *(f32x4*)(A + o) = av;
    *(f32x4*)(B + o) = -2.0f * a * bl * su;
}

// Phase 2: sequential combine over the 128 chunks (one lane per state elem).
// Overwrites Q with chunk-start means and A with chunk-start variances,
// and writes the final-state tail of the output.
__global__ __launch_bounds__(256) void ema_phase2(
    const float* __restrict__ h_a0, const float* __restrict__ h_sd0,
    const float* __restrict__ alpha,
    float* __restrict__ QSa, float* __restrict__ ASv,
    const float* __restrict__ B, float* __restrict__ out)
{
    const int i = blockIdx.x * blockDim.x + threadIdx.x;   // 0..STATE-1
    if (i >= STATE) return;
    const float a = alpha[i];
    const float b = 1.0f - a;
    float bl = b;
    #pragma unroll
    for (int s = 0; s < 6; ++s) bl *= bl;                  // b^CHUNK
    const float cc = bl * (1.0f - bl);                     // C = b^L (1 - b^L)

    float Ha = h_a0[i];
    const float sd = h_sd0[i];
    float Hv = sd * sd;
    for (int c = 0; c < NCHUNK; ++c) {
        const size_t o = (size_t)c * STATE + i;
        const float q  = QSa[o];
        const float av = ASv[o];
        const float bb = B[o];
        QSa[o] = Ha;                                       // start state of chunk c
        ASv[o] = Hv;
        const float HaN = bl * Ha + q;
        const float HvN = bl * Hv + av + bb * Ha + cc * Ha * Ha;
        Ha = HaN;
        Hv = fmaxf(HvN, 0.0f);                             // guard fp cancellation
    }
    const size_t tail = (size_t)T_STEPS * ROW;
    out[tail + i]         = Ha;                            // hN_a
    out[tail + STATE + i] = sqrtf(Hv);                     // sqrt(hN_v)
}

// Phase 3: replay each chunk exactly like the reference and stream outputs.
__global__ __launch_bounds__(BLOCK_T) void ema_phase3(
    const float* __restrict__ x, const float* __restrict__ alpha,
    const float* __restrict__ Sa, const float* __restrict__ Sv,
    float* __restrict__ out)
{
    __shared__ __align__(16) float sx[CHUNK * DIM];
    const int c   = blockIdx.x;
    const int tx  = threadIdx.x, ty = threadIdx.y;
    const int tid = ty * BLOCK_X + tx;
    stage_chunk(x + (size_t)c * CHUNK * DIM, sx, tid);

    const int n = blockIdx.y * BLOCK_Y + ty;
    const int i = n * DIM + tx * 4;
    const f32x4 a = *(const f32x4*)(alpha + i);
    const f32x4 b = 1.0f - a;
    const size_t o = (size_t)c * STATE + i;
    f32x4 Ha = *(const f32x4*)(Sa + o);
    f32x4 Hv = *(const f32x4*)(Sv + o);

    size_t oa = (size_t)c * CHUNK * ROW + (size_t)n * DIM + tx * 4;
    #pragma unroll 4
    for (int k = 0; k < CHUNK; ++k) {
        const f32x4 xk = *(const f32x4*)(sx + k * DIM + tx * 4);
        const f32x4 dd = xk - Ha;
        Hv = b * (Hv + a * dd * dd);
        Ha = Ha + a * dd;
        f32x4 sv;
        sv.x = sqrtf(Hv.x); sv.y = sqrtf(Hv.y);
        sv.z = sqrtf(Hv.z); sv.w = sqrtf(Hv.w);
        __builtin_nontemporal_store(Ha, (f32x4*)(out + oa));          // out_a row
        __builtin_nontemporal_store(sv, (f32x4*)(out + oa + STATE));  // sqrt(out_v)
        oa += ROW;
    }
}

// Deterministic fallback if workspace is too small: direct scan, 4096 lanes.
__global__ __launch_bounds__(DIM) void ema_fallback(
    const float* __restrict__ x, const float* __restrict__ h_a0,
    const float* __restrict__ h_sd0, const float* __restrict__ alpha,
    float* __restrict__ out)
{
    const int d = threadIdx.x;        // 0..127
    const int n = blockIdx.x;         // 0..31
    const int i = n * DIM + d;
    const float a = alpha[i], b = 1.0f - a;
    float Ha = h_a0[i];
    const float sd = h_sd0[i];
    float Hv = sd * sd;
    size_t oa = (size_t)n * DIM + d;
    for (int t = 0; t < T_STEPS; ++t) {
        const float xk = x[t * DIM + d];
        const float dd = xk - Ha;
        Hv = b * (Hv + a * dd * dd);
        Ha += a * dd;
        __builtin_nontemporal_store(Ha, out + oa);
        __builtin_nontemporal_store(sqrtf(Hv), out + oa + STATE);
        oa += ROW;
    }
    const size_t tail = (size_t)T_STEPS * ROW;
    out[tail + i]         = Ha;
    out[tail + STATE + i] = sqrtf(Hv);
}

extern "C" void kernel_launch(void* const* d_in, const int* /*in_sizes*/, int /*n_in*/,
                              void* d_out, int /*out_size*/, void* d_ws, size_t ws_size,
                              hipStream_t stream)
{
    const float* x     = (const float*)d_in[0];
    const float* h_a0  = (const float*)d_in[1];
    const float* h_sd0 = (const float*)d_in[2];
    const float* alpha = (const float*)d_in[3];
    float* out = (float*)d_out;

    const size_t need = (size_t)3 * NCHUNK * STATE * sizeof(float);  // 6 MB
    if (ws_size < need) {
        ema_fallback<<<dim3(NH), dim3(DIM), 0, stream>>>(x, h_a0, h_sd0, alpha, out);
        return;
    }

    float* Q = (float*)d_ws;                       // -> chunk-start means after ph2
    float* A = Q + (size_t)NCHUNK * STATE;         // -> chunk-start vars  after ph2
    float* B = A + (size_t)NCHUNK * STATE;

    dim3 blk(BLOCK_X, BLOCK_Y);                    // 256 threads
    dim3 grd(NCHUNK, NH / BLOCK_Y);                // 128 x 4 blocks
    ema_phase1<<<grd, blk, 0, stream>>>(x, alpha, Q, A, B);
    ema_phase2<<<dim3(STATE / 256), dim3(256), 0, stream>>>(h_a0, h_sd0, alpha, Q, A, B, out);
    ema_phase3<<<grd, blk, 0, stream>>>(x, alpha, Q, A, out);
}